// Inception1d_residual_62680752718396
// MI455X (gfx1250) — compile-verified
//
#include <hip/hip_runtime.h>

typedef __attribute__((ext_vector_type(2))) float f32x2;
typedef __attribute__((ext_vector_type(8))) float f32x8;

#define B_SZ    64
#define C_IN    36
#define L_SZ    16384
#define KW      15              // real max kernel width
#define KWP     16              // padded window: tap 15 always zero weight
#define PAD     7
#define KG      (C_IN * KWP)    // 576 = padded GEMM K dimension
#define MT      48              // M padded to 3 WMMA tiles of 16
#define NTILE   256             // output positions per workgroup (8 waves x 32)
#define XSTRIDE 272             // NTILE + halo (max col 255+15=270) padded to 272

// Packed-weight LDS/global layout (one contiguous f32x2 per A fragment):
//   flat = (((c*4 + t4)*2 + half)*MT + m)*2 + v,  tap t = t4*4 + half*2 + v
#define WPACK_ELEMS (C_IN * 4 * 2 * MT * 2)   // 27648 floats = 110,592 B

// ---------------------------------------------------------------------------
// Kernel 1: pack all branch weights (plus folded +1/36 channel-mean on the
// center tap) into the fragment-ordered layout above. Rows m>=36 and tap 15
// are zero padding.
// ---------------------------------------------------------------------------
__global__ __launch_bounds__(256) void pack_weights_kernel(
    const float* __restrict__ w1,
    const float* __restrict__ w3,  const float* __restrict__ w5,
    const float* __restrict__ w7,  const float* __restrict__ w9,
    const float* __restrict__ w11, const float* __restrict__ w13,
    const float* __restrict__ w15,
    float* __restrict__ wpack) {
  int idx = blockIdx.x * 256 + threadIdx.x;
  if (idx >= WPACK_ELEMS) return;

  const float* wk[7] = {w3, w5, w7, w9, w11, w13, w15};

  int v    = idx & 1;
  int rest = idx >> 1;
  int m    = rest % MT;
  rest     = rest / MT;
  int half = rest & 1;
  int g    = rest >> 1;         // g = c*4 + t4
  int c    = g >> 2;
  int t4   = g & 3;
  int t    = t4 * 4 + half * 2 + v;   // tap in padded 16-window

  float val = 0.0f;
  if (m < C_IN && t < KW) {
    if (m == 0) {                       // 1-tap branch: w1 [1, C, 1]
      if (t == PAD) val = w1[c];
    } else {
      int mb = m - 1;
      int br = mb / 5;                  // branch 0..6 -> K = 3,5,...,15
      int o  = mb - br * 5;
      int K  = 3 + 2 * br;
      int j  = t - (PAD - K / 2);       // tap inside this branch's kernel
      if (j >= 0 && j < K) val = wk[br][(o * C_IN + c) * K + j];
    }
    if (t == PAD) val += (1.0f / (float)C_IN);  // folded channel-mean residual
  }
  wpack[idx] = val;
}

// ---------------------------------------------------------------------------
// Kernel 2: GEMM  out[48(ch) x 256(pos)] = W[576 x 48]^T * im2col(x)[576 x 256]
// per (batch, position-tile) workgroup via V_WMMA_F32_16X16X4_F32.
// 8 waves; each wave: 2 N-subtiles x 3 M-tiles = 6 accumulators.
// Outer loop over channels (scalar), inner 4 k-steps fully unrolled with
// immediate LDS offsets; A fragments are single ds_load_b64 each.
// ---------------------------------------------------------------------------
__global__ __launch_bounds__(256) void inception_wmma_kernel(
    const float* __restrict__ x,
    const float* __restrict__ wpack,
    float* __restrict__ out) {
  extern __shared__ float lds[];
  float* xs  = lds;                     // [C_IN][XSTRIDE]
  float* wsm = lds + C_IN * XSTRIDE;    // WPACK_ELEMS, fragment-ordered

  const int tid = threadIdx.x;
  const int b   = blockIdx.x >> 6;            // L_SZ / NTILE == 64 tiles
  const int n0  = (blockIdx.x & 63) * NTILE;

  // Stage x tile with halo (zero outside [0, L) == conv zero padding).
  const float* xb = x + (size_t)b * C_IN * L_SZ;
  for (int c = 0; c < C_IN; ++c) {
    for (int j = tid; j < XSTRIDE; j += 256) {
      int l = n0 - PAD + j;
      float v = 0.0f;
      if (l >= 0 && l < L_SZ) v = xb[c * L_SZ + l];
      xs[c * XSTRIDE + j] = v;
    }
  }
  // Stage packed weights as b128 copies (hot in L2 across all workgroups).
  {
    const float4* src = (const float4*)wpack;
    float4*       dst = (float4*)wsm;
    for (int i = tid; i < WPACK_ELEMS / 4; i += 256) dst[i] = src[i];
  }
  __syncthreads();

  const int lane = tid & 31;
  const int wv   = tid >> 5;            // 0..7 -> 32-position slab
  const int half = lane >> 4;           // 0/1: which half-wave
  const int lm   = lane & 15;
  const int col0 = wv * 32 + lm;        // N-subtile 0 position
  const int col1 = col0 + 16;           // N-subtile 1 position

  f32x8 acc00 = {}, acc01 = {}, acc02 = {};   // subtile 0, M-tiles 0..2
  f32x8 acc10 = {}, acc11 = {}, acc12 = {};   // subtile 1, M-tiles 0..2

  // Per-lane base pointers; all loop-carried updates are constant strides.
  const float* xrow0 = xs + col0 + 2 * half;       // + c*XSTRIDE per channel
  const float* xrow1 = xs + col1 + 2 * half;
  const float* wrow  = wsm + (half * MT + lm) * 2; // + 768 floats per channel

  for (int c = 0; c < C_IN; ++c) {
#pragma unroll
    for (int t4 = 0; t4 < 4; ++t4) {
      // B fragments (4x16 f32): lane -> N = lm, K = 2*half + v
      f32x2 bf0, bf1;
      bf0[0] = xrow0[t4 * 4 + 0];
      bf0[1] = xrow0[t4 * 4 + 1];
      bf1[0] = xrow1[t4 * 4 + 0];
      bf1[1] = xrow1[t4 * 4 + 1];
      // A fragments (16x4 f32): lane -> M = lm, K = 2*half + v; v-contiguous.
      f32x2 a0 = *(const f32x2*)(wrow + t4 * 192 + 0);
      f32x2 a1 = *(const f32x2*)(wrow + t4 * 192 + 32);
      f32x2 a2 = *(const f32x2*)(wrow + t4 * 192 + 64);

      acc00 = __builtin_amdgcn_wmma_f32_16x16x4_f32(false, a0, false, bf0,
                                                    (short)0, acc00, false, false);
      acc10 = __builtin_amdgcn_wmma_f32_16x16x4_f32(false, a0, false, bf1,
                                                    (short)0, acc10, false, false);
      acc01 = __builtin_amdgcn_wmma_f32_16x16x4_f32(false, a1, false, bf0,
                                                    (short)0, acc01, false, false);
      acc11 = __builtin_amdgcn_wmma_f32_16x16x4_f32(false, a1, false, bf1,
                                                    (short)0, acc11, false, false);
      acc02 = __builtin_amdgcn_wmma_f32_16x16x4_f32(false, a2, false, bf0,
                                                    (short)0, acc02, false, false);
      acc12 = __builtin_amdgcn_wmma_f32_16x16x4_f32(false, a2, false, bf1,
                                                    (short)0, acc12, false, false);
    }
    xrow0 += XSTRIDE;
    xrow1 += XSTRIDE;
    wrow  += 4 * 2 * MT * 2;  // 768 floats per channel
  }

  // Epilogue: C/D layout is VGPR r, lanes0-15 -> M=r, lanes16-31 -> M=8+r.
  float* ob0 = out + (size_t)b * C_IN * L_SZ + n0 + col0;
  float* ob1 = ob0 + 16;
#pragma unroll
  for (int r = 0; r < 8; ++r) {
    int ch0 = r + half * 8;             // M-tile 0: channels 0..15 (always valid)
    ob0[(size_t)ch0 * L_SZ]        = acc00[r];
    ob1[(size_t)ch0 * L_SZ]        = acc10[r];
    ob0[(size_t)(ch0 + 16) * L_SZ] = acc01[r];  // M-tile 1: channels 16..31
    ob1[(size_t)(ch0 + 16) * L_SZ] = acc11[r];
    int ch2 = ch0 + 32;                 // M-tile 2: only 32..35 are real
    if (ch2 < C_IN) {
      ob0[(size_t)ch2 * L_SZ] = acc02[r];
      ob1[(size_t)ch2 * L_SZ] = acc12[r];
    }
  }
}

// ---------------------------------------------------------------------------
extern "C" void kernel_launch(void* const* d_in, const int* in_sizes, int n_in,
                              void* d_out, int out_size, void* d_ws, size_t ws_size,
                              hipStream_t stream) {
  (void)in_sizes; (void)n_in; (void)out_size; (void)ws_size;

  const float* x   = (const float*)d_in[0];
  const float* w1  = (const float*)d_in[1];
  const float* w3  = (const float*)d_in[2];
  const float* w5  = (const float*)d_in[3];
  const float* w7  = (const float*)d_in[4];
  const float* w9  = (const float*)d_in[5];
  const float* w11 = (const float*)d_in[6];
  const float* w13 = (const float*)d_in[7];
  const float* w15 = (const float*)d_in[8];

  float* wpack = (float*)d_ws;          // WPACK_ELEMS floats = 110,592 B
  float* out   = (float*)d_out;

  pack_weights_kernel<<<(WPACK_ELEMS + 255) / 256, 256, 0, stream>>>(
      w1, w3, w5, w7, w9, w11, w13, w15, wpack);

  const size_t shmem = (size_t)(C_IN * XSTRIDE + WPACK_ELEMS) * sizeof(float); // 149,760 B
  (void)hipFuncSetAttribute((const void*)inception_wmma_kernel,
                            hipFuncAttributeMaxDynamicSharedMemorySize,
                            (int)shmem);

  inception_wmma_kernel<<<B_SZ * (L_SZ / NTILE), 256, shmem, stream>>>(
      x, wpack, out);
}